// VariationalQuantumLayer_34626026341166
// MI455X (gfx1250) — compile-verified
//
#include <hip/hip_runtime.h>

// ============================================================================
// Variational quantum layer, MI455X (gfx1250, wave32).
//
// One wave per batch element; 10-qubit state (1024 complex64 amps) lives
// entirely in VGPRs: 4 "tiles" x 8 VGPRs x (re,im) = 64 VGPRs.
//
// Amplitude index i (bit9..bit0) = (q0 q1 ... q9), q0 = MSB:
//   row   m = (q0,q1,q2,q3) -> WMMA C/D layout: vgpr = m&7, lane = n + 16*(m>>3)
//   tile  t = (q4,q5)       -> which 8-VGPR tile
//   col   n = (q6,q7,q8,q9) -> lane bits 3:0
//
// Per-layer work:
//   q0-q3 RX : fused into one 16x16 complex tensor unitary, applied as
//              D = U * S_tile with V_WMMA_F32_16X16X4_F32 (fp32 matrix cores,
//              D layout == state layout, results land in place).
//   q4,q5 RX : register-pair VALU gates across tiles (no lane crossing).
//   q6-q9 RX : lane-pair gates via DPP permutes (pure VALU, no counters):
//              xor1=quad_perm 0xB1, xor2=quad_perm 0x4E,
//              xor4=row_xmask 0x164, xor8=row_xmask 0x168.
//   CNOT ring: compile-time register renames, v_cndmask swaps, DPP, and a
//              ds_swizzle SWAPX16 only where the half-wave boundary is crossed.
//
// The initial RY(x) layer on |0..0> is NOT simulated gate-by-gate: it yields
// a product state amp(i) = prod_q (bit_q(i) ? sin(x_q/2) : cos(x_q/2)),
// materialized directly in ~60 VALU ops.
// ============================================================================

typedef __attribute__((ext_vector_type(2))) float v2f;
typedef __attribute__((ext_vector_type(8))) float v8f;

#define DEVINL __device__ __forceinline__

// DPP controls
#define DPP_XOR1 0xB1   // quad_perm [1,0,3,2]
#define DPP_XOR2 0x4E   // quad_perm [2,3,0,1]
#define DPP_XOR4 0x164  // row_xmask:4
#define DPP_XOR8 0x168  // row_xmask:8
// ds_swizzle imm for SWAPX16 (group-of-32: xor=0x10, or=0, and=0x1f)
#define SWZ_XOR16 0x401F

DEVINL v8f wmma4(v2f a, v2f b, v8f c) {
  return __builtin_amdgcn_wmma_f32_16x16x4_f32(false, a, false, b, (short)0, c,
                                               false, false);
}

template <int CTRL>
DEVINL float dppx(float v) {
  int i = __float_as_int(v);
  return __int_as_float(__builtin_amdgcn_update_dpp(i, i, CTRL, 0xF, 0xF, true));
}

DEVINL float swiz16(float v) {
  return __int_as_float(__builtin_amdgcn_ds_swizzle(__float_as_int(v), SWZ_XOR16));
}

DEVINL void swapf(float& a, float& b) { float t = a; a = b; b = t; }

DEVINL void cswap(bool p, float& a, float& b) {
  float t = a;
  a = p ? b : a;
  b = p ? t : b;
}

// RX on a register pair: a0' = c a0 - i s a1 ; a1' = -i s a0 + c a1.
DEVINL void rx_pair(float& r0, float& i0, float& r1, float& i1, float c, float s) {
  float t0r = r0, t0i = i0, t1r = r1, t1i = i1;
  r0 = c * t0r + s * t1i;  i0 = c * t0i - s * t1r;
  r1 = c * t1r + s * t0i;  i1 = c * t1i - s * t0r;
}

// RX where the paired bit is a lane bit (same formula both sides -- no select):
//   re' = c*re + s*im_partner ; im' = c*im - s*re_partner
template <int CTRL>
DEVINL void rx_lane_dpp(float& r, float& i, float c, float s) {
  float pr = dppx<CTRL>(r);
  float pi = dppx<CTRL>(i);
  r = c * r + s * pi;
  i = c * i - s * pr;
}

// Extract B-operand fragment (chunk rows mlo (lanes<16) / mlo+2 (lanes>=16))
// from a C-layout tile: one ds_swizzle SWAPX16 + one select after folding.
DEVINL float bfrag(const float S[8], int mlo, int lane) {
  int mhi = mlo + 2;
  float lo = S[mlo & 7];
  if ((mlo >> 3) != 0) lo = swiz16(lo);
  float hi = S[mhi & 7];
  if ((mhi >> 3) != 1) hi = swiz16(hi);
  return (lane < 16) ? lo : hi;
}

// Build A-layout fragments of U = RX(q0) (x) RX(q1) (x) RX(q2) (x) RX(q3):
// entry(m,k) = (-i)^popc(m^k) * prod_j (diff_j ? s_j : c_j).
// Each lane holds entries (m, k), m = lane&15, k = 4c + 2*(lane>>4) + b.
DEVINL void build_rx_u(float ARe[8], float AIm[8], const float* c4,
                       const float* s4, int lane) {
  int m = lane & 15, khalf = lane >> 4;
#pragma unroll
  for (int c = 0; c < 4; ++c)
#pragma unroll
    for (int b = 0; b < 2; ++b) {
      int k = 4 * c + 2 * khalf + b;
      int d = m ^ k;
      float f = ((d & 8) ? s4[0] : c4[0]) * ((d & 4) ? s4[1] : c4[1]) *
                ((d & 2) ? s4[2] : c4[2]) * ((d & 1) ? s4[3] : c4[3]);
      int p = __popc(d) & 3;
      ARe[2 * c + b] = (p == 0) ? f : ((p == 2) ? -f : 0.f);
      AIm[2 * c + b] = (p == 1) ? -f : ((p == 3) ? f : 0.f);
    }
}

// D = U * S on every tile. Complex matmul = 4 real WMMA chains:
//   Dre = URe*Sre - UIm*Sim ; Dim = URe*Sim + UIm*Sre
// (minus folded into a negated A fragment; f32 NEG bits are C-negate only).
DEVINL void apply_row_gate(float sr[4][8], float si[4][8], const float ARe[8],
                           const float AIm[8], int lane) {
#pragma unroll
  for (int t = 0; t < 4; ++t) {
    float Bre[8], Bim[8];
#pragma unroll
    for (int c = 0; c < 4; ++c)
#pragma unroll
      for (int j = 0; j < 2; ++j) {
        Bre[2 * c + j] = bfrag(sr[t], 4 * c + j, lane);
        Bim[2 * c + j] = bfrag(si[t], 4 * c + j, lane);
      }
    v8f dre = {0.f, 0.f, 0.f, 0.f, 0.f, 0.f, 0.f, 0.f};
    v8f dim = {0.f, 0.f, 0.f, 0.f, 0.f, 0.f, 0.f, 0.f};
#pragma unroll
    for (int c = 0; c < 4; ++c) {
      v2f are = {ARe[2 * c], ARe[2 * c + 1]};
      v2f aim = {AIm[2 * c], AIm[2 * c + 1]};
      v2f ain = {-AIm[2 * c], -AIm[2 * c + 1]};
      v2f bre = {Bre[2 * c], Bre[2 * c + 1]};
      v2f bim = {Bim[2 * c], Bim[2 * c + 1]};
      dre = wmma4(are, bre, dre);
      dre = wmma4(ain, bim, dre);
      dim = wmma4(are, bim, dim);
      dim = wmma4(aim, bre, dim);
    }
#pragma unroll
    for (int i = 0; i < 8; ++i) { sr[t][i] = dre[i]; si[t][i] = dim[i]; }
  }
}

// CNOT with ctrl = lane bit cm, tgt = lane-xor DPP permute CTRL.
template <int CTRL>
DEVINL void cnot_lane_dpp(float sr[4][8], float si[4][8], int cm, int lane) {
  bool p = (lane & cm) != 0;
#pragma unroll
  for (int t = 0; t < 4; ++t)
#pragma unroll
    for (int v = 0; v < 8; ++v) {
      float a = dppx<CTRL>(sr[t][v]);
      sr[t][v] = p ? a : sr[t][v];
      float b = dppx<CTRL>(si[t][v]);
      si[t][v] = p ? b : si[t][v];
    }
}

// Ring of CNOT(q, q+1 mod 10). All pure permutations in this layout.
DEVINL void cnot_chain(float sr[4][8], float si[4][8], int lane) {
  // (0,1): ctrl lane bit4 (q0), tgt vgpr bit2 (q1) -> conditional reg swaps
  bool c0 = (lane & 16) != 0;
#pragma unroll
  for (int t = 0; t < 4; ++t)
#pragma unroll
    for (int v = 0; v < 4; ++v) {
      cswap(c0, sr[t][v], sr[t][v | 4]);
      cswap(c0, si[t][v], si[t][v | 4]);
    }
  // (1,2): ctrl v bit2, tgt v bit1 -> compile-time swaps (4,6),(5,7)
#pragma unroll
  for (int t = 0; t < 4; ++t) {
    swapf(sr[t][4], sr[t][6]); swapf(si[t][4], si[t][6]);
    swapf(sr[t][5], sr[t][7]); swapf(si[t][5], si[t][7]);
  }
  // (2,3): ctrl v bit1, tgt v bit0 -> swaps (2,3),(6,7)
#pragma unroll
  for (int t = 0; t < 4; ++t) {
    swapf(sr[t][2], sr[t][3]); swapf(si[t][2], si[t][3]);
    swapf(sr[t][6], sr[t][7]); swapf(si[t][6], si[t][7]);
  }
  // (3,4): ctrl v bit0 (q3), tgt t bit1 (q4) -> for odd v swap (t0,t2),(t1,t3)
#pragma unroll
  for (int v = 1; v < 8; v += 2) {
    swapf(sr[0][v], sr[2][v]); swapf(si[0][v], si[2][v]);
    swapf(sr[1][v], sr[3][v]); swapf(si[1][v], si[3][v]);
  }
  // (4,5): ctrl t bit1, tgt t bit0 -> swap tiles 2,3
#pragma unroll
  for (int v = 0; v < 8; ++v) {
    swapf(sr[2][v], sr[3][v]); swapf(si[2][v], si[3][v]);
  }
  // (5,6): ctrl t bit0 (q5), tgt lane bit3 (q6) -> DPP xor8 on tiles 1,3
#pragma unroll
  for (int v = 0; v < 8; ++v) {
    sr[1][v] = dppx<DPP_XOR8>(sr[1][v]);
    si[1][v] = dppx<DPP_XOR8>(si[1][v]);
    sr[3][v] = dppx<DPP_XOR8>(sr[3][v]);
    si[3][v] = dppx<DPP_XOR8>(si[3][v]);
  }
  cnot_lane_dpp<DPP_XOR4>(sr, si, 8, lane);  // (6,7)
  cnot_lane_dpp<DPP_XOR2>(sr, si, 4, lane);  // (7,8)
  cnot_lane_dpp<DPP_XOR1>(sr, si, 2, lane);  // (8,9)
  // (9,0): ctrl q9 = lane bit0, tgt q0 = lane bit4 -> ds_swizzle SWAPX16
  {
    bool p9 = (lane & 1) != 0;
#pragma unroll
    for (int t = 0; t < 4; ++t) {
      float a[8], b[8];
#pragma unroll
      for (int v = 0; v < 8; ++v) {  // batch the DS ops, then select
        a[v] = swiz16(sr[t][v]);
        b[v] = swiz16(si[t][v]);
      }
#pragma unroll
      for (int v = 0; v < 8; ++v) {
        sr[t][v] = p9 ? a[v] : sr[t][v];
        si[t][v] = p9 ? b[v] : si[t][v];
      }
    }
  }
}

DEVINL float wave_sum(float v) {
  v += dppx<DPP_XOR1>(v);
  v += dppx<DPP_XOR2>(v);
  v += dppx<DPP_XOR4>(v);
  v += dppx<DPP_XOR8>(v);
  v += swiz16(v);
  return v;
}

__global__ __launch_bounds__(256) void vql_kernel(const float* __restrict__ x,
                                                  const float* __restrict__ w,
                                                  float* __restrict__ out) {
  const int lane = threadIdx.x & 31;
  const int b = blockIdx.x * (blockDim.x >> 5) + (threadIdx.x >> 5);

  // per-batch input half-angles
  float cx[10], sx[10];
#pragma unroll
  for (int q = 0; q < 10; ++q) {
    float th = 0.5f * x[b * 10 + q];
    cx[q] = __cosf(th);
    sx[q] = __sinf(th);
  }

  // ---- RY(x) layer on |0..0> == product state, built directly ----
  float sr[4][8], si[4][8];
  {
    float lanef = ((lane & 8) ? sx[6] : cx[6]) * ((lane & 4) ? sx[7] : cx[7]) *
                  ((lane & 2) ? sx[8] : cx[8]) * ((lane & 1) ? sx[9] : cx[9]);
    float g = lanef * ((lane & 16) ? sx[0] : cx[0]);  // q0 = lane-half bit
    float rowf[8];
#pragma unroll
    for (int v = 0; v < 8; ++v)
      rowf[v] = ((v & 4) ? sx[1] : cx[1]) * ((v & 2) ? sx[2] : cx[2]) *
                ((v & 1) ? sx[3] : cx[3]);
    float tf[4];
#pragma unroll
    for (int t = 0; t < 4; ++t)
      tf[t] = ((t & 2) ? sx[4] : cx[4]) * ((t & 1) ? sx[5] : cx[5]);
#pragma unroll
    for (int t = 0; t < 4; ++t)
#pragma unroll
      for (int v = 0; v < 8; ++v) {
        sr[t][v] = g * rowf[v] * tf[t];
        si[t][v] = 0.f;
      }
  }

  // ---- entangling layers: RX(w[l,q]) on all qubits, then CNOT ring ----
  for (int l = 0; l < 4; ++l) {
    float cw[10], sw[10];
#pragma unroll
    for (int q = 0; q < 10; ++q) {
      float th = 0.5f * w[l * 10 + q];
      cw[q] = __cosf(th);
      sw[q] = __sinf(th);
    }
    {  // RX q0..q3 fused -> WMMA left-multiply per tile
      float ARe[8], AIm[8];
      float c4[4] = {cw[0], cw[1], cw[2], cw[3]};
      float s4[4] = {sw[0], sw[1], sw[2], sw[3]};
      build_rx_u(ARe, AIm, c4, s4, lane);
      apply_row_gate(sr, si, ARe, AIm, lane);
    }
#pragma unroll
    for (int v = 0; v < 8; ++v) {  // RX q4: tile pairs (0,2),(1,3)
      rx_pair(sr[0][v], si[0][v], sr[2][v], si[2][v], cw[4], sw[4]);
      rx_pair(sr[1][v], si[1][v], sr[3][v], si[3][v], cw[4], sw[4]);
    }
#pragma unroll
    for (int v = 0; v < 8; ++v) {  // RX q5: tile pairs (0,1),(2,3)
      rx_pair(sr[0][v], si[0][v], sr[1][v], si[1][v], cw[5], sw[5]);
      rx_pair(sr[2][v], si[2][v], sr[3][v], si[3][v], cw[5], sw[5]);
    }
    // RX q6..q9: DPP lane gates (masks 8,4,2,1)
#pragma unroll
    for (int t = 0; t < 4; ++t)
#pragma unroll
      for (int v = 0; v < 8; ++v) {
        rx_lane_dpp<DPP_XOR8>(sr[t][v], si[t][v], cw[6], sw[6]);
        rx_lane_dpp<DPP_XOR4>(sr[t][v], si[t][v], cw[7], sw[7]);
        rx_lane_dpp<DPP_XOR2>(sr[t][v], si[t][v], cw[8], sw[8]);
        rx_lane_dpp<DPP_XOR1>(sr[t][v], si[t][v], cw[9], sw[9]);
      }
    cnot_chain(sr, si, lane);
  }

  // ---- measurement: z_q = sum_i sign_q(i) * |amp_i|^2 ----
  float sAll = 0.f, z1 = 0.f, z2 = 0.f, z3 = 0.f, z4 = 0.f, z5 = 0.f;
#pragma unroll
  for (int t = 0; t < 4; ++t)
#pragma unroll
    for (int v = 0; v < 8; ++v) {
      float p = sr[t][v] * sr[t][v] + si[t][v] * si[t][v];
      sAll += p;
      z1 += (v & 4) ? -p : p;  // q1 = m bit2
      z2 += (v & 2) ? -p : p;  // q2 = m bit1
      z3 += (v & 1) ? -p : p;  // q3 = m bit0
      z4 += (t & 2) ? -p : p;  // q4 = t bit1
      z5 += (t & 1) ? -p : p;  // q5 = t bit0
    }
  float z[10];
  z[0] = (lane & 16) ? -sAll : sAll;  // q0 = m bit3 = lane bit4
  z[1] = z1; z[2] = z2; z[3] = z3; z[4] = z4; z[5] = z5;
  z[6] = (lane & 8) ? -sAll : sAll;   // q6..q9 = lane bits 3..0
  z[7] = (lane & 4) ? -sAll : sAll;
  z[8] = (lane & 2) ? -sAll : sAll;
  z[9] = (lane & 1) ? -sAll : sAll;
#pragma unroll
  for (int q = 0; q < 10; ++q) z[q] = wave_sum(z[q]);

  if (lane == 0) {
#pragma unroll
    for (int q = 0; q < 10; ++q) out[b * 10 + q] = z[q];
  }
}

extern "C" void kernel_launch(void* const* d_in, const int* in_sizes, int n_in,
                              void* d_out, int out_size, void* d_ws,
                              size_t ws_size, hipStream_t stream) {
  const float* x = (const float*)d_in[0];  // (2048, 10) float32
  const float* w = (const float*)d_in[1];  // (4, 10) float32
  float* out = (float*)d_out;              // (2048, 10) float32
  const int batch = in_sizes[0] / 10;      // 2048
  const int wavesPerBlock = 8;             // 256 threads = 8 wave32
  dim3 block(32 * wavesPerBlock);
  dim3 grid((batch + wavesPerBlock - 1) / wavesPerBlock);
  vql_kernel<<<grid, block, 0, stream>>>(x, w, out);
}